// RoPE_90847148245015
// MI455X (gfx1250) — compile-verified
//
#include <hip/hip_runtime.h>
#include <cstdint>

// RoPE applied as the sparse 2x2 block rotation it actually is.
// out[b,2p,  h,w] = cos*f[b,2p,h,w] - sin*f[b,2p+1,h,w]
// out[b,2p+1,h,w] = sin*f[b,2p,h,w] + cos*f[b,2p+1,h,w]
// cos = R[h,2p,2p], sin = R[h,2p+1,2p]  (dense R is [H, C, C], row-major)

typedef float v4f __attribute__((ext_vector_type(4)));

#define B_   16
#define C_   256
#define H_   128
#define W_   128
#define NPAIR (C_/2)      // 128 rotation pairs
#define HW_  (H_*W_)      // 16384 (channel stride in elements)

__global__ __launch_bounds__(256) void rope_sparse_kernel(
    const float* __restrict__ feats,
    const float* __restrict__ rota,
    float* __restrict__ out)
{
    const int h   = blockIdx.x;   // position
    const int b   = blockIdx.y;   // batch
    const int tid = threadIdx.x;  // 0..255

    // coef[0..127]   = cos(h * theta_p)
    // coef[128..255] = sin(h * theta_p)
    __shared__ float coef[2 * NPAIR];

    // ---- Stage coefficients via CDNA5 async global->LDS (ASYNCcnt path) ----
    // Gather from the dense R matrix: cos at (2p)*C + 2p, sin at (2p+1)*C + 2p.
    {
        const int p     = tid & (NPAIR - 1);
        const int isSin = tid >> 7;                       // 0 for tid<128, 1 else
        const uint32_t voff = (uint32_t)(((2 * p + isSin) * C_ + 2 * p) * 4);
        const uint64_t sbase =
            (uint64_t)(uintptr_t)(rota + (size_t)h * C_ * C_);
        // Generic (flat) pointer into LDS: low 32 bits == LDS byte offset.
        const uint32_t ldsoff = (uint32_t)(uintptr_t)(&coef[tid]);
        asm volatile("global_load_async_to_lds_b32 %0, %1, %2"
                     :: "v"(ldsoff), "v"(voff), "s"(sbase)
                     : "memory");
        asm volatile("s_wait_asynccnt 0" ::: "memory");
    }
    __syncthreads();

    // ---- Streaming rotation: wave owns pairs, lane owns a float4 of w ----
    const int wave = tid >> 5;        // 0..7
    const int lane = tid & 31;        // 0..31
    const int w4   = lane * 4;        // 4 floats per lane -> covers W=128

    const size_t slab = (size_t)b * C_ * HW_ + (size_t)h * W_ + (size_t)w4;
    const float* __restrict__ src = feats + slab;
    float* __restrict__ dst = out + slab;

#pragma unroll 4
    for (int it = 0; it < NPAIR / 8; ++it) {
        const int p = wave + it * 8;          // pair index for whole wave
        const float cv = coef[p];             // ds_load broadcast
        const float sv = coef[NPAIR + p];

        const size_t off0 = (size_t)(2 * p) * HW_;
        const size_t off1 = off0 + HW_;

        // Streamed once: non-temporal loads/stores (TH=NT) to bypass caches.
        v4f x = __builtin_nontemporal_load((const v4f*)(src + off0));
        v4f y = __builtin_nontemporal_load((const v4f*)(src + off1));

        v4f r0 = cv * x - sv * y;
        v4f r1 = sv * x + cv * y;

        __builtin_nontemporal_store(r0, (v4f*)(dst + off0));
        __builtin_nontemporal_store(r1, (v4f*)(dst + off1));
    }
}

extern "C" void kernel_launch(void* const* d_in, const int* in_sizes, int n_in,
                              void* d_out, int out_size, void* d_ws, size_t ws_size,
                              hipStream_t stream) {
    (void)in_sizes; (void)n_in; (void)out_size; (void)d_ws; (void)ws_size;
    const float* feats = (const float*)d_in[0];  // [16,256,128,128] f32
    const float* rota  = (const float*)d_in[1];  // [128,256,256]    f32
    float* out = (float*)d_out;                  // [16,256,128,128] f32

    dim3 grid(H_, B_);   // 128 x 16 = 2048 blocks
    dim3 block(256);     // 8 waves (wave32)
    rope_sparse_kernel<<<grid, block, 0, stream>>>(feats, rota, out);
}